// GCN_8555574853994
// MI455X (gfx1250) — compile-verified
//
#include <hip/hip_runtime.h>

// ---------------------------------------------------------------------------
// GCN on MI455X (gfx1250, wave32).
//  - Dense projections: V_WMMA_F32_16X16X4_F32 (full fp32 precision) with both
//    operand tiles staged into LDS by the Tensor Data Mover (TDM), using the
//    D# pad feature for bank-conflict-free LDS strides.
//  - Graph aggregation: coalesced gathers + global f32 atomics (L2-resident:
//    feature matrices are 51MB/25MB << 192MB L2), with global_prefetch of the
//    gather row 64 edges ahead.
// ---------------------------------------------------------------------------

typedef __attribute__((ext_vector_type(2))) float v2f;
typedef __attribute__((ext_vector_type(8))) float v8f;
typedef __attribute__((ext_vector_type(4))) unsigned int v4u;
typedef __attribute__((ext_vector_type(8))) int v8i;
typedef __attribute__((ext_vector_type(4))) int v4i;

#define N_NODES 50000
#define N_EDGES 800000
#define IN_F 256
#define H_F 256
#define C_F 128
#define K_DIM 256

#define LDSA_STRIDE 258   // 256 + 2 pad DWORDs  (TDM pad_interval=256, pad_amount=2)
#define LDSB_STRIDE 129   // 128 + 1 pad DWORD   (TDM pad_interval=128, pad_amount=1)
#define K_CHUNK 64

// ---------------- TDM 2-D tile load: global -> LDS ----------------
// D# per CDNA5 ISA 8.3/8.4: group0 = {count|flags, lds_addr, global_addr, type=2},
// group1 = {mask/data_size/pad, dims, tile dims, stride}. Groups 2/3 zero (2-D).
// This toolchain (clang-23 / therock-10.0) exposes the 6-arg builtin form.
__device__ __forceinline__ void tdm_load_2d(unsigned ldsOff, const void* gptr,
                                            unsigned tile_w, unsigned tile_h,
                                            unsigned long long row_stride,
                                            unsigned pad_iv_enc, unsigned pad_amt_enc,
                                            bool pad_en) {
    unsigned long long ga = (unsigned long long)(uintptr_t)gptr;
    v4u g0 = { 1u,                                   // count=1 (valid user descriptor)
               ldsOff,                               // lds_addr (bytes)
               (unsigned)(ga & 0xFFFFFFFFu),         // global_addr[31:0]
               (unsigned)((ga >> 32) & 0x01FFFFFFu) | (2u << 30) };  // addr[56:32] | type=2
    unsigned flags = (2u << 16);                     // data_size = 2 -> 4 bytes
    if (pad_en) flags |= (1u << 20) | (pad_iv_enc << 22) | (pad_amt_enc << 25);
    v8i g1;
    g1[0] = (int)flags;
    g1[1] = (int)((tile_w & 0xFFFFu) << 16);                          // tensor_dim0[15:0]
    g1[2] = (int)((tile_w >> 16) | ((tile_h & 0xFFFFu) << 16));       // dim0[31:16]|dim1[15:0]
    g1[3] = (int)((tile_h >> 16) | ((tile_w & 0xFFFFu) << 16));       // dim1[31:16]|tile_dim0
    g1[4] = (int)(tile_h & 0xFFFFu);                                  // tile_dim1
    g1[5] = (int)(row_stride & 0xFFFFFFFFull);                        // dim0_stride[31:0]
    g1[6] = (int)((row_stride >> 32) & 0xFFFFull);                    // dim0_stride[47:32]
    g1[7] = 0;
    v4i z4 = { 0, 0, 0, 0 };
    v8i z8 = { 0, 0, 0, 0, 0, 0, 0, 0 };
    __builtin_amdgcn_tensor_load_to_lds(g0, g1, z4, z4, z8, 0);
}

// ---------------- utility kernels ----------------

__global__ void gcn_zero(float* __restrict__ p, long long n) {
    long long i = (long long)blockIdx.x * blockDim.x + threadIdx.x;
    if (i < n) p[i] = 0.0f;
}

__global__ void gcn_degrees(const int* __restrict__ src, const int* __restrict__ dst,
                            float* __restrict__ degO, float* __restrict__ degI, int E) {
    int e = blockIdx.x * blockDim.x + threadIdx.x;
    if (e < E) {
        atomicAdd(&degO[src[e]], 1.0f);
        atomicAdd(&degI[dst[e]], 1.0f);
    }
}

__global__ void gcn_norms(float* __restrict__ degO, float* __restrict__ degI, int n) {
    int i = blockIdx.x * blockDim.x + threadIdx.x;
    if (i < n) {
        degO[i] = rsqrtf(fmaxf(degO[i], 1.0f));
        degI[i] = rsqrtf(fmaxf(degI[i], 1.0f));
    }
}

// ---------------- WMMA GEMM:  C[M,N] = (A[M,K] * rowscale) @ B[K,N] ----------------
// Block = 256 threads (8 waves) -> one 16-row tile, 128 columns (8 x 16x16 tiles).
// Wave 0 TDM-stages the 16xK A tile once and the B tile in K-chunks of 64 rows;
// every wave then runs a pure LDS + v_wmma_f32_16x16x4_f32 inner loop.
template <int N>
__global__ void gcn_gemm_wmma(const float* __restrict__ A,
                              const float* __restrict__ B,
                              const float* __restrict__ rowscale,
                              float* __restrict__ C) {
    __shared__ float ldsA[16 * LDSA_STRIDE + 4];        // 16 rows x K, padded
    __shared__ float ldsB[K_CHUNK * LDSB_STRIDE + 4];   // 64 rows x 128 cols, padded

    const int lane   = threadIdx.x & 31;
    const int wave   = threadIdx.x >> 5;
    const int laneLo = lane & 15;
    const int laneHi = lane >> 4;

    const int row0    = blockIdx.x * 16;
    const int colBase = blockIdx.y * 128;

    // Stage A tile (16 x 256, row stride K) -> LDS rows padded to 258 DWORDs.
    if (threadIdx.x < 32) {
        tdm_load_2d((unsigned)(uintptr_t)(void*)ldsA, A + (size_t)row0 * K_DIM,
                    K_DIM, 16, K_DIM, /*iv: 256 dw*/ 7, /*amt: 2 dw*/ 1, true);
        __builtin_amdgcn_s_wait_tensorcnt(0);
    }

    const float ns  = rowscale[row0 + laneLo];          // lanes L and L+16 share a row
    const int  bcol = wave * 16 + laneLo;

    v8f acc = {};
    for (int kc = 0; kc < K_DIM; kc += K_CHUNK) {
        // Stage B chunk (64 x 128, row stride N) -> LDS rows padded to 129 DWORDs.
        if (threadIdx.x < 32) {
            tdm_load_2d((unsigned)(uintptr_t)(void*)ldsB, B + (size_t)kc * N + colBase,
                        128, K_CHUNK, N, /*iv: 128 dw*/ 6, /*amt: 1 dw*/ 0, true);
            __builtin_amdgcn_s_wait_tensorcnt(0);
        }
        __syncthreads();

#pragma unroll 4
        for (int kk = 0; kk < K_CHUNK; kk += 4) {
            const int kl = kk + 2 * laneHi;
            v2f a = *(const v2f*)(ldsA + laneLo * LDSA_STRIDE + kc + kl);
            a.x *= ns;
            a.y *= ns;
            v2f b;
            b.x = ldsB[kl * LDSB_STRIDE + bcol];
            b.y = ldsB[(kl + 1) * LDSB_STRIDE + bcol];
            acc = __builtin_amdgcn_wmma_f32_16x16x4_f32(
                /*neg_a=*/false, a, /*neg_b=*/false, b,
                /*c_mod=*/(short)0, acc, /*reuse_a=*/false, /*reuse_b=*/false);
        }
        __syncthreads();   // protect ldsB before next chunk overwrite
    }

    // C/D layout: VGPR r -> row r + 8*laneHi, col = laneLo.
    float* Cp = C + (size_t)(row0 + 8 * laneHi) * N + colBase + wave * 16 + laneLo;
#pragma unroll
    for (int r = 0; r < 8; ++r)
        Cp[(size_t)r * N] = acc[r];
}

// ---------------- edge scatter-add:  AGG[dst[e], :] += H[src[e], :] ----------------
// Each wave handles a 32-feature slice of one edge -> src/dst wave-uniform,
// H reads and AGG atomics fully coalesced. Prefetch the gather row 64 edges
// ahead into L2 while the atomics drain.
template <int F>
__global__ void gcn_scatter(const float* __restrict__ H,
                            const int* __restrict__ src,
                            const int* __restrict__ dst,
                            float* __restrict__ AGG, int E) {
    long long tid = (long long)blockIdx.x * blockDim.x + threadIdx.x;
    if (tid >= (long long)E * F) return;
    const int e = (int)(tid / F);
    const int f = (int)(tid % F);
    const int s = src[e];
    const int d = dst[e];
    if (e + 64 < E) {
        const int sp = src[e + 64];
        __builtin_prefetch(&H[(size_t)sp * F + f], 0, 0);
    }
    atomicAdd(&AGG[(size_t)d * F + f], H[(size_t)s * F + f]);
}

// ---------------- epilogue: OUT = act(AGG * norm_dst[row] + bias) ----------------
template <int F, bool RELU>
__global__ void gcn_epilogue(const float* __restrict__ AGG,
                             const float* __restrict__ nd,
                             const float* __restrict__ bias,
                             float* __restrict__ OUT, int Nn) {
    long long tid = (long long)blockIdx.x * blockDim.x + threadIdx.x;
    if (tid >= (long long)Nn * F) return;
    const int i = (int)(tid / F);
    const int f = (int)(tid % F);
    float v = AGG[tid] * nd[i] + bias[f];
    OUT[tid] = RELU ? fmaxf(v, 0.0f) : v;
}

// ---------------- launcher ----------------

extern "C" void kernel_launch(void* const* d_in, const int* in_sizes, int n_in,
                              void* d_out, int out_size, void* d_ws, size_t ws_size,
                              hipStream_t stream) {
    const float* x   = (const float*)d_in[0];
    const int*   src = (const int*)d_in[1];
    const int*   dst = (const int*)d_in[2];
    const float* W1  = (const float*)d_in[3];
    const float* b1  = (const float*)d_in[4];
    const float* W2  = (const float*)d_in[5];
    const float* b2  = (const float*)d_in[6];
    float*       out = (float*)d_out;

    float* ws       = (float*)d_ws;
    float* norm_src = ws;                           // 50000 (deg_out -> norm, in place)
    float* norm_dst = ws + 50176;                   // 50000 (deg_in  -> norm, in place)
    float* h1       = ws + 100352;                  // 50000*256
    float* agg1     = h1 + (size_t)N_NODES * H_F;   // 50000*256; reused as h2

    const int T = 256;
    const long long nh1 = (long long)N_NODES * H_F;   // 12.8M
    const long long no  = (long long)N_NODES * C_F;   // 6.4M

    // degrees -> norms
    gcn_zero<<<(100352 + T - 1) / T, T, 0, stream>>>(ws, 100352);
    gcn_degrees<<<(N_EDGES + T - 1) / T, T, 0, stream>>>(src, dst, norm_src, norm_dst, N_EDGES);
    gcn_norms<<<(N_NODES + T - 1) / T, T, 0, stream>>>(norm_src, norm_dst, N_NODES);

    // layer 1: h1 = (x * norm_src) @ W1 ; agg1 = scatter ; h1 = relu(agg1*norm_dst + b1)
    gcn_gemm_wmma<H_F><<<dim3(N_NODES / 16, H_F / 128), T, 0, stream>>>(x, W1, norm_src, h1);
    gcn_zero<<<(int)((nh1 + T - 1) / T), T, 0, stream>>>(agg1, nh1);
    gcn_scatter<H_F><<<(int)(((long long)N_EDGES * H_F + T - 1) / T), T, 0, stream>>>(h1, src, dst, agg1, N_EDGES);
    gcn_epilogue<H_F, true><<<(int)((nh1 + T - 1) / T), T, 0, stream>>>(agg1, norm_dst, b1, h1, N_NODES);

    // layer 2: h2 = (h1 * norm_src) @ W2 (into agg1) ; out = scatter ; out = out*norm_dst + b2
    gcn_gemm_wmma<C_F><<<dim3(N_NODES / 16, C_F / 128), T, 0, stream>>>(h1, W2, norm_src, agg1);
    gcn_zero<<<(int)((no + T - 1) / T), T, 0, stream>>>(out, no);
    gcn_scatter<C_F><<<(int)(((long long)N_EDGES * C_F + T - 1) / T), T, 0, stream>>>(agg1, src, dst, out, N_EDGES);
    gcn_epilogue<C_F, false><<<(int)((no + T - 1) / T), T, 0, stream>>>(out, norm_dst, b2, out, N_NODES);
}